// DGL_GCN_33062658244614
// MI455X (gfx1250) — compile-verified
//
#include <hip/hip_runtime.h>

typedef float v2f __attribute__((ext_vector_type(2)));
typedef float v8f __attribute__((ext_vector_type(8)));

#define NFEAT 128

// ---------------- utility kernels ----------------

__global__ void gcn_zero(float* __restrict__ p, long long n) {
  long long i = (long long)blockIdx.x * blockDim.x + threadIdx.x;
  long long stride = (long long)gridDim.x * blockDim.x;
  for (; i < n; i += stride) p[i] = 0.0f;
}

__global__ void gcn_count_deg(const int* __restrict__ src, const int* __restrict__ dst,
                              float* __restrict__ dout, float* __restrict__ din, int E) {
  int e = blockIdx.x * blockDim.x + threadIdx.x;
  if (e >= E) return;
  atomicAdd(&dout[src[e]], 1.0f);
  atomicAdd(&din[dst[e]], 1.0f);
}

__global__ void gcn_rsqrt_clip(float* __restrict__ p, int n) {
  int i = blockIdx.x * blockDim.x + threadIdx.x;
  if (i < n) p[i] = rsqrtf(fmaxf(p[i], 1.0f));
}

// ---------------- WMMA fp32 GEMM: out[n,:] = (x[n,:]*s[n]) @ W ----------------
// Block = 256 threads = 8 waves; each wave computes a 16-node x 128-feature tile.
// W staged in LDS as float2 K-pairs: row kp holds {W[2kp][n], W[2kp+1][n]}.
// Odd kp rows store columns bit4-XOR-swizzled (n^16) so the two half-waves of a
// b64 fragment load hit disjoint 32-bank groups (conflict-free ds_load_b64).

// one full MMA round: 8 output tiles, same A operand -> reuse_a on tiles 1..7
#define MMA_ROUND(A, B)                                                                                     \
  acc[0] = __builtin_amdgcn_wmma_f32_16x16x4_f32(false, A, false, B[0], (short)0, acc[0], false, false);    \
  acc[1] = __builtin_amdgcn_wmma_f32_16x16x4_f32(false, A, false, B[1], (short)0, acc[1], true, false);     \
  acc[2] = __builtin_amdgcn_wmma_f32_16x16x4_f32(false, A, false, B[2], (short)0, acc[2], true, false);     \
  acc[3] = __builtin_amdgcn_wmma_f32_16x16x4_f32(false, A, false, B[3], (short)0, acc[3], true, false);     \
  acc[4] = __builtin_amdgcn_wmma_f32_16x16x4_f32(false, A, false, B[4], (short)0, acc[4], true, false);     \
  acc[5] = __builtin_amdgcn_wmma_f32_16x16x4_f32(false, A, false, B[5], (short)0, acc[5], true, false);     \
  acc[6] = __builtin_amdgcn_wmma_f32_16x16x4_f32(false, A, false, B[6], (short)0, acc[6], true, false);     \
  acc[7] = __builtin_amdgcn_wmma_f32_16x16x4_f32(false, A, false, B[7], (short)0, acc[7], true, false);

#define LOADB(DST, K)                                                      \
  {                                                                        \
    const v2f* wk = Wp + ((((K) >> 1) + half) << 7) + l15;                 \
    _Pragma("unroll") for (int n0 = 0; n0 < 8; ++n0)                       \
        DST[n0] = wk[(n0 << 4) ^ hbit];                                    \
  }

__global__ __launch_bounds__(256) void gcn_gemm_wmma(const float* __restrict__ x,
                                                     const float* __restrict__ rout,
                                                     const float* __restrict__ W,
                                                     float* __restrict__ out,
                                                     int nnodes) {
  __shared__ v2f Wp[64 * 128];  // 64 KB
  const int tid = threadIdx.x;
  for (int i = tid; i < 64 * 128; i += 256) {
    const int kp = i >> 7;
    const int n  = i & 127;
    v2f pr;
    pr.x = W[(2 * kp) * NFEAT + n];
    pr.y = W[(2 * kp + 1) * NFEAT + n];
    Wp[(kp << 7) + ((kp & 1) ? (n ^ 16) : n)] = pr;
  }
  __syncthreads();

  const int wave = tid >> 5;
  const int lane = tid & 31;
  const int half = lane >> 4;   // 0: lanes 0-15 (K=k,k+1), 1: lanes 16-31 (K=k+2,k+3)
  const int l15  = lane & 15;
  const int hbit = half << 4;
  const int node0 = blockIdx.x * 128 + wave * 16;
  if (node0 >= nnodes) return;  // tiles are all-or-nothing (40000 % 16 == 0)

  const int row = node0 + l15;
  const float s = rout[row];
  const float* xrow = x + (long long)row * NFEAT + half * 2;

  v8f acc[8] = {};
  v2f a0, a1;
  v2f b0[8], b1[8];

  // prologue: stage k = 0
  {
    float2 av = *(const float2*)(xrow);
    a0.x = av.x * s;
    a0.y = av.y * s;
  }
  LOADB(b0, 0)

#pragma unroll
  for (int k = 0; k < NFEAT; k += 8) {
    // prefetch k+4 (always in range: k+4 <= 124)
    {
      float2 av = *(const float2*)(xrow + k + 4);
      a1.x = av.x * s;
      a1.y = av.y * s;
    }
    LOADB(b1, k + 4)
    MMA_ROUND(a0, b0)
    if (k + 8 < NFEAT) {  // folds at full unroll
      float2 av = *(const float2*)(xrow + k + 8);
      a0.x = av.x * s;
      a0.y = av.y * s;
      LOADB(b0, k + 8)
    }
    MMA_ROUND(a1, b1)
  }

  // C/D layout: VGPR r -> M = r (lanes 0-15) or M = r+8 (lanes 16-31), N = l15
#pragma unroll
  for (int n0 = 0; n0 < 8; ++n0) {
#pragma unroll
    for (int r = 0; r < 8; ++r) {
      const int m = half * 8 + r;
      out[(long long)(node0 + m) * NFEAT + n0 * 16 + l15] = acc[n0][r];
    }
  }
}

// ---------------- edge scatter-add: agg[dst] += h[src], one wave per edge ----------------

__global__ void gcn_scatter(const float* __restrict__ h, const int* __restrict__ src,
                            const int* __restrict__ dst, float* __restrict__ agg, int E) {
  int wid = (blockIdx.x * blockDim.x + threadIdx.x) >> 5;
  int lane = threadIdx.x & 31;
  if (wid >= E) return;
  int s = __builtin_amdgcn_readfirstlane(__builtin_nontemporal_load(&src[wid]));
  int d = __builtin_amdgcn_readfirstlane(__builtin_nontemporal_load(&dst[wid]));
  float4 v = *(const float4*)(h + (long long)s * NFEAT + lane * 4);
  float* p = agg + (long long)d * NFEAT + lane * 4;
  atomicAdd(p + 0, v.x);
  atomicAdd(p + 1, v.y);
  atomicAdd(p + 2, v.z);
  atomicAdd(p + 3, v.w);
}

// ---------------- dst-norm + bias + relu ----------------

__global__ void gcn_finalize(const float* __restrict__ agg, const float* __restrict__ rin,
                             const float* __restrict__ bias, float* __restrict__ out,
                             int nnodes) {
  int i = blockIdx.x * blockDim.x + threadIdx.x;
  if (i >= nnodes * NFEAT) return;
  int n = i >> 7;
  int c = i & (NFEAT - 1);
  float v = agg[i] * rin[n] + bias[c];
  out[i] = fmaxf(v, 0.0f);
}

// ---------------- launch ----------------

extern "C" void kernel_launch(void* const* d_in, const int* in_sizes, int n_in,
                              void* d_out, int out_size, void* d_ws, size_t ws_size,
                              hipStream_t stream) {
  const float* x    = (const float*)d_in[0];
  const int*   esrc = (const int*)d_in[1];
  const int*   edst = (const int*)d_in[2];
  const float* W1   = (const float*)d_in[3];
  const float* b1   = (const float*)d_in[4];
  const float* W2   = (const float*)d_in[5];
  const float* b2   = (const float*)d_in[6];

  const int N = in_sizes[0] / NFEAT;
  const int E = in_sizes[1];

  float* rout = (float*)d_ws;          // N floats
  float* rin  = rout + N;              // N floats
  size_t off = (((size_t)2 * N * sizeof(float)) + 255) & ~(size_t)255;
  float* bufA = (float*)((char*)d_ws + off);   // N*128 floats
  float* out  = (float*)d_out;                 // N*128 floats (also used as temp)

  const dim3 blk(256);
  const long long total = (long long)N * NFEAT;
  const dim3 gElem((unsigned)((total + 255) / 256));
  const dim3 gDeg((unsigned)((2 * N + 255) / 256));
  const dim3 gEdge((unsigned)((E + 255) / 256));
  const dim3 gGemm((unsigned)((N + 127) / 128));
  const dim3 gScat((unsigned)((E + 7) / 8));

  // degrees -> rsqrt(max(deg,1))
  gcn_zero<<<gDeg, blk, 0, stream>>>(rout, (long long)2 * N);
  gcn_count_deg<<<gEdge, blk, 0, stream>>>(esrc, edst, rout, rin, E);
  gcn_rsqrt_clip<<<gDeg, blk, 0, stream>>>(rout, 2 * N);

  // layer 1: bufA = (x * rout) @ W1 ; out = scatter(bufA) ; bufA = relu(out*rin + b1)
  gcn_gemm_wmma<<<gGemm, blk, 0, stream>>>(x, rout, W1, bufA, N);
  gcn_zero<<<dim3(1024), blk, 0, stream>>>(out, total);
  gcn_scatter<<<gScat, blk, 0, stream>>>(bufA, esrc, edst, out, E);
  gcn_finalize<<<gElem, blk, 0, stream>>>(out, rin, b1, bufA, N);

  // layer 2: out = (bufA * rout) @ W2 ; bufA = scatter(out) ; out = relu(bufA*rin + b2)
  gcn_gemm_wmma<<<gGemm, blk, 0, stream>>>(bufA, rout, W2, out, N);
  gcn_zero<<<dim3(1024), blk, 0, stream>>>(bufA, total);
  gcn_scatter<<<gScat, blk, 0, stream>>>(out, esrc, edst, bufA, E);
  gcn_finalize<<<gElem, blk, 0, stream>>>(bufA, rin, b2, out, N);
}